// LSTMClassifier_46471546143153
// MI455X (gfx1250) — compile-verified
//
#include <hip/hip_runtime.h>
#include <hip/hip_bf16.h>

// Problem dims (reference): B=64, S=512, E=512, H=512, V=32000, T=2, L=1
#define S_   512
#define E_   512
#define H_   512
#define G_   2048      // 4*H
#define BATCH 64
#define NBLK  16       // scan workgroups (one per WGP, W_hh slice fits in 320KB LDS)

typedef __attribute__((ext_vector_type(2))) float v2f;
typedef __attribute__((ext_vector_type(8))) float v8f;

// workspace byte offsets
#define X_OFF   0u                      // 512*512 f32 = 1 MB
#define XG_OFF  (1u << 20)              // 512*2048 f32 = 4 MB
#define HS_OFF  (5u << 20)              // 512*512 f32 = 1 MB
#define HG_OFF  (6u << 20)              // 512 f32 h-exchange
#define BAR_OFF ((6u << 20) + 4096u)    // 2 u32 barrier state

// ---------------------------------------------------------------------------
// Phase 1a: gather embedding rows for batch sample 63 -> X[512,512]
// ---------------------------------------------------------------------------
__global__ void gather_x_kernel(const int* __restrict__ sentence,
                                const float* __restrict__ emb,
                                float* __restrict__ X) {
  const int t = blockIdx.x;                       // timestep
  const int idx = sentence[(BATCH - 1) * S_ + t]; // token id of sample 63
  const float4* src = (const float4*)(emb + (size_t)idx * E_);
  float4* dst = (float4*)(X + (size_t)t * E_);
  dst[threadIdx.x] = src[threadIdx.x];            // blockDim.x == E_/4 == 128
}

// ---------------------------------------------------------------------------
// Phase 1b: XG[512,2048] = X @ W_ih^T + (b_ih + b_hh), fp32 WMMA 16x16x4.
// One 16x16 output tile per wave; 8 waves per block.
// A (16x4 f32): lanes 0-15 -> K=0,1 ; lanes 16-31 -> K=2,3 (2 VGPRs/lane)
// B (4x16 f32): row k striped across lanes within a VGPR (mirror of A)
// C/D (16x16 f32): VGPR r -> M=r (lanes 0-15) / M=8+r (lanes 16-31)
// ---------------------------------------------------------------------------
__global__ void xg_gemm_wmma_kernel(const float* __restrict__ X,
                                    const float* __restrict__ W_ih,
                                    const float* __restrict__ b_ih,
                                    const float* __restrict__ b_hh,
                                    float* __restrict__ XG) {
  const int wave = threadIdx.x >> 5;
  const int lane = threadIdx.x & 31;
  const int half = lane >> 4;        // 0: lanes 0-15, 1: lanes 16-31
  const int l    = lane & 15;

  const int tile = blockIdx.x * 8 + wave;   // 4096 tiles total
  const int tm = tile >> 7;                 // 0..31  (M tiles over t)
  const int tn = tile & 127;                // 0..127 (N tiles over gates)
  const int row0 = tm * 16;
  const int col0 = tn * 16;

  const float* arow = X    + (size_t)(row0 + l) * E_;  // X[m, :]
  const float* brow = W_ih + (size_t)(col0 + l) * E_;  // W_ih[n, :] == B[:, n]

  v8f acc = {};
  #pragma unroll 4
  for (int k0 = 0; k0 < E_; k0 += 4) {
    const int k = k0 + 2 * half;
    v2f a, b;
    a.x = arow[k]; a.y = arow[k + 1];
    b.x = brow[k]; b.y = brow[k + 1];
    acc = __builtin_amdgcn_wmma_f32_16x16x4_f32(
        /*neg_a=*/false, a, /*neg_b=*/false, b,
        /*c_mod=*/(short)0, acc, /*reuse_a=*/false, /*reuse_b=*/false);
  }

  const float bias = b_ih[col0 + l] + b_hh[col0 + l];
  #pragma unroll
  for (int r = 0; r < 8; ++r) {
    const int row = row0 + half * 8 + r;
    XG[(size_t)row * G_ + col0 + l] = acc[r] + bias;
  }
}

// ---------------------------------------------------------------------------
// Lightweight sense-reversing grid barrier (NBLK resident blocks).
// ---------------------------------------------------------------------------
__device__ __forceinline__ void grid_sync(unsigned* count,
                                          volatile unsigned* phase,
                                          unsigned nblk) {
  __syncthreads();
  if (threadIdx.x == 0) {
    __threadfence();
    const unsigned my = *phase;
    if (atomicAdd(count, 1u) == nblk - 1u) {
      *count = 0u;
      __threadfence();
      atomicAdd((unsigned*)phase, 1u);
    } else {
      while (*phase == my) { __builtin_amdgcn_s_sleep(2); }
    }
  }
  __syncthreads();
}

// ---------------------------------------------------------------------------
// Phase 2: sequential LSTM scan, batch row 63 only.
// Block b owns h-slots [b*32, b*32+32) and their 4 gate rows (i,f,g,o):
// 128 rows x 512 f32 = 256 KB of W_hh staged in LDS (320 KB/WGP on CDNA5)
// via the CDNA5 async global->LDS engine (ASYNCcnt), no VGPR round-trip.
// Per step: LDS-resident matvec, local gate math, 2 KB h exchange via L2
// (h broadcast also pulled with async-to-LDS at device scope).
//
// NOTE: this kernel has no static __shared__, so the dynamic-LDS segment
// starts at LDS byte offset 0 => LDS addresses for the async VDST operand
// are plain byte offsets into smem[].
// ---------------------------------------------------------------------------
__global__ void lstm_scan_kernel(const float* __restrict__ XG,
                                 const float* __restrict__ W_hh,
                                 const float* __restrict__ h0,
                                 const float* __restrict__ c0,
                                 float* __restrict__ HS,
                                 float* hglob,
                                 unsigned* bar) {
  extern __shared__ float smem[];
  float* Wsh = smem;                 // 128 * 512 floats (256 KB), LDS offset 0
  float* hsh = smem + 128 * H_;      // 512
  float* csh = hsh + H_;             // 32
  float* gsh = csh + 32;             // 128

  const unsigned HSH_BYTE_OFF = (unsigned)(128 * H_ * sizeof(float));

  const int tid = threadIdx.x;       // 0..127
  const int b   = blockIdx.x;        // 0..15
  const int q   = tid >> 5;          // gate (i/f/g/o)
  const int s   = tid & 31;          // slot within block's 32 h-slots
  const int grow = q * H_ + b * 32 + s;   // global gate row

  // Stage this block's 128 W_hh rows straight into LDS via async DMA:
  // each thread issues 128 x b128 memory->LDS transfers for its row.
  {
    const float* gsrc = W_hh + (size_t)grow * H_;
    const unsigned lds_row = (unsigned)(tid * H_ * sizeof(float));
    #pragma unroll 4
    for (int i = 0; i < H_ / 4; ++i) {
      asm volatile("global_load_async_to_lds_b128 %0, %1, off"
                   :: "v"(lds_row + (unsigned)(i * 16)), "v"(gsrc + i * 4)
                   : "memory");
    }
  }
  // init h (full vector) and c (own slots) for batch row 63
  ((float4*)hsh)[tid] = ((const float4*)(h0 + (size_t)(BATCH - 1) * H_))[tid];
  if (tid < 32) csh[tid] = c0[(size_t)(BATCH - 1) * H_ + b * 32 + tid];
  asm volatile("s_wait_asynccnt 0x0" ::: "memory");  // W slice resident in LDS
  __syncthreads();

  unsigned* count = bar;
  volatile unsigned* phase = bar + 1;

  for (int t = 0; t < S_; ++t) {
    // gate pre-activation for row `grow`: xg + W_hh[row] . h  (all from LDS)
    {
      const float4* w  = (const float4*)(Wsh + (size_t)tid * H_);
      const float4* hv = (const float4*)hsh;
      float acc = XG[(size_t)t * G_ + grow];
      #pragma unroll 8
      for (int i = 0; i < H_ / 4; ++i) {
        const float4 wv = w[i];
        const float4 hh = hv[i];
        acc += wv.x * hh.x + wv.y * hh.y + wv.z * hh.z + wv.w * hh.w;
      }
      gsh[tid] = acc;
    }
    __syncthreads();

    if (tid < 32) {   // torch gate order: i, f, g, o
      float iv = gsh[tid];
      float fv = gsh[32 + tid];
      float gv = gsh[64 + tid];
      float ov = gsh[96 + tid];
      iv = 1.f / (1.f + __expf(-iv));
      fv = 1.f / (1.f + __expf(-fv));
      gv = tanhf(gv);
      ov = 1.f / (1.f + __expf(-ov));
      const float cn = fv * csh[tid] + iv * gv;
      const float hn = ov * tanhf(cn);
      csh[tid] = cn;
      hglob[b * 32 + tid] = hn;                   // publish own h slots
      HS[(size_t)t * H_ + b * 32 + tid] = hn;     // record h_t (sample 63)
      __threadfence();
    }
    grid_sync(count, phase, NBLK);   // all h slots published

    // pull full updated h into LDS: one async b128 per thread, device scope
    // so the transfer is served from device-coherent L2 (not a stale L0).
    {
      const float* gaddr = hglob + tid * 4;
      const unsigned lds_off = HSH_BYTE_OFF + (unsigned)(tid * 16);
      asm volatile("global_load_async_to_lds_b128 %0, %1, off scope:SCOPE_DEV"
                   :: "v"(lds_off), "v"(gaddr)
                   : "memory");
      asm volatile("s_wait_asynccnt 0x0" ::: "memory");
    }
    grid_sync(count, phase, NBLK);   // everyone consumed h before next write
  }
}

// ---------------------------------------------------------------------------
// Phase 3: out[t,k] = HS[t,:] . W_lin[k,:] + b_lin[k]   (T = 2)
// ---------------------------------------------------------------------------
__global__ void final_linear_kernel(const float* __restrict__ HS,
                                    const float* __restrict__ W_lin,
                                    const float* __restrict__ b_lin,
                                    float* __restrict__ out) {
  const int id = blockIdx.x * blockDim.x + threadIdx.x;  // 0..1023
  if (id >= S_ * 2) return;
  const int t = id >> 1;
  const int k = id & 1;
  const float4* hv = (const float4*)(HS + (size_t)t * H_);
  const float4* wv = (const float4*)(W_lin + (size_t)k * H_);
  float acc = 0.f;
  #pragma unroll 8
  for (int i = 0; i < H_ / 4; ++i) {
    const float4 a = hv[i];
    const float4 w = wv[i];
    acc += a.x * w.x + a.y * w.y + a.z * w.z + a.w * w.w;
  }
  out[id] = acc + b_lin[k];
}

// ---------------------------------------------------------------------------
extern "C" void kernel_launch(void* const* d_in, const int* in_sizes, int n_in,
                              void* d_out, int out_size, void* d_ws, size_t ws_size,
                              hipStream_t stream) {
  const int*   sentence = (const int*)  d_in[0];
  const float* h0       = (const float*)d_in[1];
  const float* c0       = (const float*)d_in[2];
  const float* emb      = (const float*)d_in[3];
  const float* W_ih     = (const float*)d_in[4];
  const float* W_hh     = (const float*)d_in[5];
  const float* b_ih     = (const float*)d_in[6];
  const float* b_hh     = (const float*)d_in[7];
  const float* W_lin    = (const float*)d_in[8];
  const float* b_lin    = (const float*)d_in[9];
  float* out = (float*)d_out;

  char* ws = (char*)d_ws;
  float*    X     = (float*)(ws + X_OFF);
  float*    XG    = (float*)(ws + XG_OFF);
  float*    HS    = (float*)(ws + HS_OFF);
  float*    hglob = (float*)(ws + HG_OFF);
  unsigned* bar   = (unsigned*)(ws + BAR_OFF);

  // reset grid-barrier state every launch (graph-capture safe)
  hipMemsetAsync(bar, 0, 2 * sizeof(unsigned), stream);

  gather_x_kernel<<<S_, E_ / 4, 0, stream>>>(sentence, emb, X);

  // 32 x 128 = 4096 tiles, 8 waves/block -> 512 blocks
  xg_gemm_wmma_kernel<<<512, 256, 0, stream>>>(X, W_ih, b_ih, b_hh, XG);

  const size_t lds_bytes = (size_t)(128 * H_ + H_ + 32 + 128) * sizeof(float);
  lstm_scan_kernel<<<NBLK, 128, lds_bytes, stream>>>(XG, W_hh, h0, c0, HS, hglob, bar);

  final_linear_kernel<<<4, 256, 0, stream>>>(HS, W_lin, b_lin, out);
}